// TeCNO_59605556134325
// MI455X (gfx1250) — compile-verified
//
#include <hip/hip_runtime.h>

// ---------------- model constants ----------------
#define EMBED 16
#define WINDOW 90
#define HIDDEN 32
#define NUM_LAYERS 10
#define NUM_CLASSES 7
#define KSIZE 3
#define BN 2
#define TN 4096
#define NROWS (BN * TN)
#define MINF (-1.0e30f)

typedef __attribute__((ext_vector_type(2))) float v2f;
typedef __attribute__((ext_vector_type(8))) float v8f;

// fp32 WMMA 16x16x4: D = A(16x4) * B(4x16) + C(16x16), exact fp32 math.
// 8-arg VOP3P form: (neg_a, A, neg_b, B, c_mod, C, reuse_a, reuse_b)
static __device__ __forceinline__ v8f wmma4(v2f a, v2f b, v8f c) {
  return __builtin_amdgcn_wmma_f32_16x16x4_f32(false, a, false, b, (short)0, c,
                                               false, false);
}

// Fragment index helpers (wave32):
//  A 16x4: lanes 0-15 hold M=lane, {K=0,1}; lanes 16-31 hold M=lane-16, {K=2,3}
//  B 4x16: lanes 0-15 hold N=lane, rows {v, }; lanes 16-31 rows {v+2}
//  C/D:    VGPR r holds row r (lanes 0-15) / row r+8 (lanes 16-31), col = lane&15

// ---------------- kernel 1: input projections + layernorm ----------------
__global__ void proj_ln_kernel(const float* __restrict__ x,
                               const float* __restrict__ lpw,
                               const float* __restrict__ lpb,
                               const float* __restrict__ lg,
                               const float* __restrict__ lb,
                               const float* __restrict__ rpw,
                               const float* __restrict__ rpb,
                               const float* __restrict__ rg,
                               const float* __restrict__ rb,
                               float* __restrict__ left,
                               float* __restrict__ right) {
  int row = blockIdx.x * blockDim.x + threadIdx.x;
  if (row >= NROWS) return;
  const float* xr = x + (size_t)row * 16;
  float hl[16], hr[16];
#pragma unroll
  for (int e = 0; e < 16; ++e) {
    float al = lpb[e], ar = rpb[e];
#pragma unroll
    for (int c = 0; c < 8; ++c) {
      al += xr[c] * lpw[e * 8 + c];
      ar += xr[8 + c] * rpw[e * 8 + c];
    }
    hl[e] = al;
    hr[e] = ar;
  }
  float mul = 0.f, mur = 0.f;
#pragma unroll
  for (int e = 0; e < 16; ++e) { mul += hl[e]; mur += hr[e]; }
  mul *= (1.0f / 16.0f);
  mur *= (1.0f / 16.0f);
  float vl = 0.f, vr = 0.f;
#pragma unroll
  for (int e = 0; e < 16; ++e) {
    float dl = hl[e] - mul, dr = hr[e] - mur;
    vl += dl * dl;
    vr += dr * dr;
  }
  vl *= (1.0f / 16.0f);
  vr *= (1.0f / 16.0f);
  float il = rsqrtf(vl + 1e-5f), ir = rsqrtf(vr + 1e-5f);
#pragma unroll
  for (int e = 0; e < 16; ++e) {
    left[(size_t)row * 16 + e] = (hl[e] - mul) * il * lg[e] + lb[e];
    right[(size_t)row * 16 + e] = (hr[e] - mur) * ir * rg[e] + rb[e];
  }
}

// ---------------- kernel 2: generic Y = X(Mx16) @ W(16x16)^T + bias ----------------
__global__ void gemm_rows16_kernel(const float* __restrict__ X,
                                   const float* __restrict__ W,
                                   const float* __restrict__ bias,
                                   float* __restrict__ Y, int M) {
  const int wave = (blockIdx.x * blockDim.x + threadIdx.x) >> 5;
  const int lane = threadIdx.x & 31;
  const int m0 = wave * 16;
  if (m0 >= M) return;  // wave-uniform
  const int n = lane & 15;
  const int half = (lane < 16) ? 0 : 8;
  const int am = lane & 15;
  const int off = (lane < 16) ? 0 : 2;
  v8f c = {0.f, 0.f, 0.f, 0.f, 0.f, 0.f, 0.f, 0.f};
  const float* xrow = X + (size_t)(m0 + am) * 16;
#pragma unroll
  for (int s = 0; s < 4; ++s) {
    v2f a;
    a.x = xrow[4 * s + off];
    a.y = xrow[4 * s + off + 1];
    v2f b;  // B[k][n] = W[n][k]
    b.x = W[n * 16 + 4 * s + off];
    b.y = W[n * 16 + 4 * s + off + 1];
    c = wmma4(a, b, c);
  }
#pragma unroll
  for (int r = 0; r < 8; ++r)
    Y[(size_t)(m0 + r + half) * 16 + n] = c[r] + bias[n];
}

// ---------------- kernel 3: windowed flash attention (one wave / 16 queries) ----
__global__ void attn_kernel(const float* __restrict__ x,
                            const float* __restrict__ Q,
                            const float* __restrict__ Km,
                            const float* __restrict__ Vm,
                            const int* __restrict__ lengths,
                            const float* __restrict__ gamma_p, int swap_bias,
                            float* __restrict__ ctx) {
  __shared__ float pbuf[16][17];
  const int qt = blockIdx.x;
  const int b = blockIdx.y;
  const int lane = threadIdx.x;
  const int q0 = qt * 16;
  int len = lengths[b];
  len = len < 1 ? 1 : (len > TN ? TN : len);
  const float gm = *gamma_p;
  const float g = logf(1.0f + expf(gm));  // softplus(gamma)

  const int n = lane & 15;
  const int half = (lane < 16) ? 0 : 8;
  const int am = lane & 15;
  const int off = (lane < 16) ? 0 : 2;

  // query-side bias coords: l_attn uses x[q,0:3], r_attn uses x[q,8:11]
  const int xqoff = swap_bias ? 8 : 0;
  float xq0[8], xq1[8], xq2[8];
#pragma unroll
  for (int r = 0; r < 8; ++r) {
    const float* p = x + (size_t)(b * TN + q0 + r + half) * 16 + xqoff;
    xq0[r] = p[0];
    xq1[r] = p[1];
    xq2[r] = p[2];
  }
  // Q fragments, pre-scaled by 1/sqrt(EMBED)
  v2f qa[4];
  const float* qrow = Q + (size_t)(b * TN + q0 + am) * 16;
#pragma unroll
  for (int s = 0; s < 4; ++s) {
    qa[s].x = qrow[4 * s + off] * 0.25f;
    qa[s].y = qrow[4 * s + off + 1] * 0.25f;
  }

  float mrow[8], lrow[8];
#pragma unroll
  for (int r = 0; r < 8; ++r) {
    mrow[r] = MINF;
    lrow[r] = 0.0f;
  }
  v8f acc = {0.f, 0.f, 0.f, 0.f, 0.f, 0.f, 0.f, 0.f};

  int ktlo = q0 - WINDOW;
  ktlo = (ktlo < 0) ? 0 : ktlo;
  ktlo >>= 4;
  for (int kt = ktlo; kt <= qt; ++kt) {
    const int k0 = kt * 16;
    // scores: S = (Q*scale) @ K^T
    v8f sv = {0.f, 0.f, 0.f, 0.f, 0.f, 0.f, 0.f, 0.f};
    const float* krow = Km + (size_t)(b * TN + k0 + n) * 16;
#pragma unroll
    for (int s = 0; s < 4; ++s) {
      v2f bf;  // B[e][n] = K[k0+n][e]
      bf.x = krow[4 * s + off];
      bf.y = krow[4 * s + off + 1];
      sv = wmma4(qa[s], bf, sv);
    }
    const int kcol = k0 + n;
    const float* xkp = x + (size_t)(b * TN + kcol) * 16 + (swap_bias ? 0 : 8);
    const float xk0 = xkp[0], xk1 = xkp[1], xk2 = xkp[2];
    const bool kv = (kcol < len);
#pragma unroll
    for (int r = 0; r < 8; ++r) {
      const int qi = q0 + r + half;
      float d0 = xq0[r] - xk0, d1 = xq1[r] - xk1, d2 = xq2[r] - xk2;
      float sval = sv[r] - g * (d0 * d0 + d1 * d1 + d2 * d2);
      bool dead = (kcol > qi) || (qi - kcol > WINDOW) || !kv;
      float val = dead ? MINF : sval;
      // row max over the 16-lane half (rows live per-half)
      float rm = val;
      rm = fmaxf(rm, __shfl_xor(rm, 1, 32));
      rm = fmaxf(rm, __shfl_xor(rm, 2, 32));
      rm = fmaxf(rm, __shfl_xor(rm, 4, 32));
      rm = fmaxf(rm, __shfl_xor(rm, 8, 32));
      float newm = fmaxf(mrow[r], rm);
      float corr = __expf(mrow[r] - newm);
      float pexp = __expf(val - newm);
      float rs = pexp;
      rs += __shfl_xor(rs, 1, 32);
      rs += __shfl_xor(rs, 2, 32);
      rs += __shfl_xor(rs, 4, 32);
      rs += __shfl_xor(rs, 8, 32);
      lrow[r] = lrow[r] * corr + rs;
      mrow[r] = newm;
      acc[r] = acc[r] * corr;
      pbuf[r + half][n] = pexp;  // stage P for A-layout re-fragmentation
    }
    __syncthreads();  // single wave; acts as LDS fence
    // acc += P @ V
#pragma unroll
    for (int s = 0; s < 4; ++s) {
      v2f pa;
      pa.x = pbuf[am][4 * s + off];
      pa.y = pbuf[am][4 * s + off + 1];
      const float* vp = Vm + (size_t)(b * TN + k0 + 4 * s + off) * 16 + n;
      v2f bf;  // B[k][e] = V[k0+k][e]
      bf.x = vp[0];
      bf.y = vp[16];
      acc = wmma4(pa, bf, acc);
    }
    __syncthreads();
  }
#pragma unroll
  for (int r = 0; r < 8; ++r)
    ctx[(size_t)(b * TN + q0 + r + half) * 16 + n] = acc[r] / lrow[r];
}

// ---------------- kernel 4: fusion + x^T residual -> h0 (B,16,T) ----------------
__global__ void fusion_kernel(const float* __restrict__ x,
                              const float* __restrict__ lctx,
                              const float* __restrict__ rctx,
                              const float* __restrict__ fw,  // 16x32
                              const float* __restrict__ fb,
                              const int* __restrict__ lengths,
                              float* __restrict__ h0) {
  const int tt = blockIdx.x;
  const int b = blockIdx.y;
  const int lane = threadIdx.x;
  const int t0 = tt * 16;
  int len = lengths[b];
  len = len < 1 ? 1 : (len > TN ? TN : len);
  const int n = lane & 15;
  const int half = (lane < 16) ? 0 : 8;
  const int am = lane & 15;
  const int off = (lane < 16) ? 0 : 2;
  const float qm = (t0 + n < len) ? 1.f : 0.f;
  v8f c = {0.f, 0.f, 0.f, 0.f, 0.f, 0.f, 0.f, 0.f};
  const size_t rowbase = (size_t)(b * TN + t0 + n) * 16;
#pragma unroll
  for (int s = 0; s < 8; ++s) {
    v2f a;
    a.x = fw[am * 32 + 4 * s + off];
    a.y = fw[am * 32 + 4 * s + off + 1];
    int cc = 4 * s + off;  // even; cc and cc+1 stay in the same half
    v2f bf;
    bf.x = (cc < 16 ? lctx[rowbase + cc] : rctx[rowbase + cc - 16]) * qm;
    bf.y = (cc + 1 < 16 ? lctx[rowbase + cc + 1] : rctx[rowbase + cc - 15]) * qm;
    c = wmma4(a, bf, c);
  }
#pragma unroll
  for (int r = 0; r < 8; ++r) {
    int e = r + half;
    h0[(size_t)(b * EMBED + e) * TN + t0 + n] =
        c[r] + fb[e] + x[(size_t)(b * TN + t0 + n) * 16 + e];
  }
}

// ---------------- kernel 5: stage input conv1x1 (Cin -> 32) ----------------
__global__ void stage_in_kernel(const float* __restrict__ hin,  // B x Cin x T
                                int Cin,
                                const float* __restrict__ w,  // 32 x Cin
                                const float* __restrict__ bias,
                                float* __restrict__ hout) {  // B x 32 x T
  const int tt = blockIdx.x;
  const int rt = blockIdx.y;
  const int b = blockIdx.z;
  const int lane = threadIdx.x;
  const int t0 = tt * 16;
  const int n = lane & 15;
  const int half = (lane < 16) ? 0 : 8;
  const int am = lane & 15;
  const int off = (lane < 16) ? 0 : 2;
  v8f c = {0.f, 0.f, 0.f, 0.f, 0.f, 0.f, 0.f, 0.f};
#pragma unroll
  for (int s = 0; s < 4; ++s) {  // K padded to 16
    int k = 4 * s + off;
    v2f a;
    a.x = (k < Cin) ? w[(rt * 16 + am) * Cin + k] : 0.f;
    a.y = (k + 1 < Cin) ? w[(rt * 16 + am) * Cin + k + 1] : 0.f;
    v2f bf;
    bf.x = (k < Cin) ? hin[(size_t)(b * Cin + k) * TN + t0 + n] : 0.f;
    bf.y = (k + 1 < Cin) ? hin[(size_t)(b * Cin + k + 1) * TN + t0 + n] : 0.f;
    c = wmma4(a, bf, c);
  }
#pragma unroll
  for (int r = 0; r < 8; ++r) {
    int ch = rt * 16 + r + half;
    hout[(size_t)(b * HIDDEN + ch) * TN + t0 + n] = c[r] + bias[ch];
  }
}

// ---------------- kernel 6: TCN layer (dilated conv + relu + 1x1 residual) ----
__global__ void tcn_layer_kernel(const float* __restrict__ hin,  // B x 32 x T
                                 const float* __restrict__ wd,   // 32x32x3
                                 const float* __restrict__ bd,
                                 const float* __restrict__ w1,  // 32x32
                                 const float* __restrict__ b1, int dil,
                                 float* __restrict__ hout) {
  __shared__ float z[HIDDEN][17];
  const int tt = blockIdx.x;
  const int b = blockIdx.y;
  const int wv = threadIdx.x >> 5;  // 2 waves: output rows [wv*16, wv*16+16)
  const int lane = threadIdx.x & 31;
  const int t0 = tt * 16;
  const int n = lane & 15;
  const int half = (lane < 16) ? 0 : 8;
  const int am = lane & 15;
  const int off = (lane < 16) ? 0 : 2;

  v8f c = {0.f, 0.f, 0.f, 0.f, 0.f, 0.f, 0.f, 0.f};
#pragma unroll
  for (int tap = 0; tap < KSIZE; ++tap) {
    const int shift = (KSIZE - 1 - tap) * dil;
    const int tsrc = t0 + n - shift;
    const bool ok = (tsrc >= 0);
#pragma unroll
    for (int s = 0; s < 8; ++s) {
      int k = 4 * s + off;
      v2f a;
      a.x = wd[((wv * 16 + am) * HIDDEN + k) * KSIZE + tap];
      a.y = wd[((wv * 16 + am) * HIDDEN + k + 1) * KSIZE + tap];
      v2f bf;
      bf.x = ok ? hin[(size_t)(b * HIDDEN + k) * TN + tsrc] : 0.f;
      bf.y = ok ? hin[(size_t)(b * HIDDEN + k + 1) * TN + tsrc] : 0.f;
      c = wmma4(a, bf, c);
    }
  }
#pragma unroll
  for (int r = 0; r < 8; ++r) {
    int ch = wv * 16 + r + half;
    z[ch][n] = fmaxf(c[r] + bd[ch], 0.f);
  }
  __syncthreads();
  v8f o = {0.f, 0.f, 0.f, 0.f, 0.f, 0.f, 0.f, 0.f};
#pragma unroll
  for (int s = 0; s < 8; ++s) {
    int k = 4 * s + off;
    v2f a;
    a.x = w1[(wv * 16 + am) * HIDDEN + k];
    a.y = w1[(wv * 16 + am) * HIDDEN + k + 1];
    v2f bf;
    bf.x = z[k][n];
    bf.y = z[k + 1][n];
    o = wmma4(a, bf, o);
  }
#pragma unroll
  for (int r = 0; r < 8; ++r) {
    int ch = wv * 16 + r + half;
    size_t idx = (size_t)(b * HIDDEN + ch) * TN + t0 + n;
    hout[idx] = hin[idx] + o[r] + b1[ch];
  }
}

// ---------------- kernel 7: stage output conv1x1 (+ optional softmax) ----------
__global__ void stage_out_kernel(const float* __restrict__ h,  // B x 32 x T
                                 const float* __restrict__ w,  // 7 x 32
                                 const float* __restrict__ bias, int do_softmax,
                                 float* __restrict__ out) {
  int idx = blockIdx.x * blockDim.x + threadIdx.x;
  if (idx >= NROWS) return;
  int b = idx / TN, t = idx - b * TN;
  float col[HIDDEN];
#pragma unroll
  for (int c = 0; c < HIDDEN; ++c) col[c] = h[(size_t)(b * HIDDEN + c) * TN + t];
  float o_[NUM_CLASSES];
#pragma unroll
  for (int o = 0; o < NUM_CLASSES; ++o) {
    float a = bias[o];
#pragma unroll
    for (int c = 0; c < HIDDEN; ++c) a += w[o * HIDDEN + c] * col[c];
    o_[o] = a;
  }
  if (do_softmax) {
    float m = o_[0];
#pragma unroll
    for (int o = 1; o < NUM_CLASSES; ++o) m = fmaxf(m, o_[o]);
    float ssum = 0.f;
#pragma unroll
    for (int o = 0; o < NUM_CLASSES; ++o) {
      o_[o] = __expf(o_[o] - m);
      ssum += o_[o];
    }
    float inv = 1.f / ssum;
#pragma unroll
    for (int o = 0; o < NUM_CLASSES; ++o)
      out[(size_t)(b * NUM_CLASSES + o) * TN + t] = o_[o] * inv;
  } else {
#pragma unroll
    for (int o = 0; o < NUM_CLASSES; ++o)
      out[((size_t)b * TN + t) * NUM_CLASSES + o] = o_[o];
  }
}

// ---------------- host orchestration ----------------
// Input index map (setup_inputs() dict insertion order, params DFS):
//  0 x, 1 lengths, 2 left_proj_w, 3 left_proj_b, 4 right_proj_w, 5 right_proj_b,
//  6 left_ln_g, 7 left_ln_b, 8 right_ln_g, 9 right_ln_b, 10 gamma,
//  11-14 l_attn{wi,bi,wo,bo}, 15-18 r_attn{wi,bi,wo,bo}, 19 fusion_w, 20 fusion_b,
//  stage s base = 21 + 44*s: w_in, b_in, 10x{wd,bd,w1,b1}, w_out, b_out
extern "C" void kernel_launch(void* const* d_in, const int* in_sizes, int n_in,
                              void* d_out, int out_size, void* d_ws,
                              size_t ws_size, hipStream_t stream) {
  (void)in_sizes;
  (void)n_in;
  (void)out_size;
  (void)ws_size;
  const float* x = (const float*)d_in[0];
  const int* lengths = (const int*)d_in[1];
  auto F = [&](int i) { return (const float*)d_in[i]; };

  float* ws = (float*)d_ws;
  size_t off = 0;
  auto alloc = [&](size_t nf) {
    float* p = ws + off;
    off += nf;
    return p;
  };
  float* left = alloc((size_t)NROWS * 16);
  float* right = alloc((size_t)NROWS * 16);
  float* Ql = alloc((size_t)NROWS * 16);
  float* Kl = alloc((size_t)NROWS * 16);
  float* Vl = alloc((size_t)NROWS * 16);
  float* Qr = alloc((size_t)NROWS * 16);
  float* Kr = alloc((size_t)NROWS * 16);
  float* Vr = alloc((size_t)NROWS * 16);
  float* cxl0 = alloc((size_t)NROWS * 16);
  float* cxr0 = alloc((size_t)NROWS * 16);
  float* cxl = alloc((size_t)NROWS * 16);
  float* cxr = alloc((size_t)NROWS * 16);
  float* h0 = alloc((size_t)BN * EMBED * TN);
  float* hA = alloc((size_t)BN * HIDDEN * TN);
  float* hB = alloc((size_t)BN * HIDDEN * TN);
  float* probs = alloc((size_t)BN * NUM_CLASSES * TN);

  // 1) projections + layernorm
  proj_ln_kernel<<<(NROWS + 255) / 256, 256, 0, stream>>>(
      x, F(2), F(3), F(6), F(7), F(4), F(5), F(8), F(9), left, right);

  // 2) QKV GEMMs (WMMA). wi is (48,16): wq rows 0-15, wk 16-31, wv 32-47.
  const float* wil = F(11);
  const float* bil = F(12);
  const float* wir = F(15);
  const float* bir = F(16);
  const int gblk = (NROWS / 16 + 7) / 8;  // 8 waves per 256-thread block
  gemm_rows16_kernel<<<gblk, 256, 0, stream>>>(left, wil, bil, Ql, NROWS);
  gemm_rows16_kernel<<<gblk, 256, 0, stream>>>(right, wil + 256, bil + 16, Kl, NROWS);
  gemm_rows16_kernel<<<gblk, 256, 0, stream>>>(right, wil + 512, bil + 32, Vl, NROWS);
  gemm_rows16_kernel<<<gblk, 256, 0, stream>>>(right, wir, bir, Qr, NROWS);
  gemm_rows16_kernel<<<gblk, 256, 0, stream>>>(left, wir + 256, bir + 16, Kr, NROWS);
  gemm_rows16_kernel<<<gblk, 256, 0, stream>>>(left, wir + 512, bir + 32, Vr, NROWS);

  // 3) windowed flash attention (WMMA), one wave per query tile
  attn_kernel<<<dim3(TN / 16, BN), 32, 0, stream>>>(x, Ql, Kl, Vl, lengths,
                                                    F(10), 0, cxl0);
  attn_kernel<<<dim3(TN / 16, BN), 32, 0, stream>>>(x, Qr, Kr, Vr, lengths,
                                                    F(10), 1, cxr0);

  // 4) attention output projections (WMMA)
  gemm_rows16_kernel<<<gblk, 256, 0, stream>>>(cxl0, F(13), F(14), cxl, NROWS);
  gemm_rows16_kernel<<<gblk, 256, 0, stream>>>(cxr0, F(17), F(18), cxr, NROWS);

  // 5) fusion + residual -> h0 (B,16,T)
  fusion_kernel<<<dim3(TN / 16, BN), 32, 0, stream>>>(x, cxl, cxr, F(19), F(20),
                                                      lengths, h0);

  // 6) two TCN stages
  for (int s = 0; s < 2; ++s) {
    const int base = 21 + 44 * s;
    const float* hin = (s == 0) ? h0 : probs;
    const int Cin = (s == 0) ? EMBED : NUM_CLASSES;
    stage_in_kernel<<<dim3(TN / 16, 2, BN), 32, 0, stream>>>(hin, Cin, F(base),
                                                             F(base + 1), hA);
    float* cur = hA;
    float* nxt = hB;
    for (int j = 0; j < NUM_LAYERS; ++j) {
      const int lb = base + 2 + 4 * j;  // wd, bd, w1, b1
      tcn_layer_kernel<<<dim3(TN / 16, BN), 64, 0, stream>>>(
          cur, F(lb), F(lb + 1), F(lb + 2), F(lb + 3), 1 << j, nxt);
      float* tmp = cur;
      cur = nxt;
      nxt = tmp;
    }
    if (s == 0) {
      stage_out_kernel<<<(NROWS + 255) / 256, 256, 0, stream>>>(
          cur, F(base + 42), F(base + 43), 1, probs);
    } else {
      stage_out_kernel<<<(NROWS + 255) / 256, 256, 0, stream>>>(
          cur, F(base + 42), F(base + 43), 0, (float*)d_out);
    }
  }
}